// TestModel_76046690943248
// MI455X (gfx1250) — compile-verified
//
#include <hip/hip_runtime.h>
#include <stdint.h>

// ---------------- problem geometry ----------------
#define ROWS    8192
#define COLS    8192
#define HALF    4096                 // output cols
#define CHUNK   2048                 // outputs per block-iteration
#define THREADS 256
#define PER_T   8                    // outputs per thread per chunk
#define DEPTH   4                    // async pipeline depth
#define GRID    2048
#define TOTAL_CHUNKS (ROWS * (HALF / CHUNK))       // 16384
#define CHUNKS_PER_BLOCK (TOTAL_CHUNKS / GRID)     // 8

static_assert(THREADS * PER_T == CHUNK, "chunk tiling");
static_assert(TOTAL_CHUNKS % GRID == 0, "even chunk split");
static_assert(CHUNKS_PER_BLOCK >= DEPTH, "pipeline fits");

#define FP8_MAX 448.0f

// ---------------- native vector types ----------------
// async-to-LDS builtin takes `int __vector(4)` pointers (per hipcc diagnostic);
// nontemporal builtins require native scalar/vector types (not HIP structs).
typedef int   v4i __attribute__((vector_size(16)));
typedef float v4f __attribute__((ext_vector_type(4)));

#define GLOBAL_AS(p) ((__attribute__((address_space(1))) v4i*)((void*)(p)))
#define LDS_AS(p)    ((__attribute__((address_space(3))) v4i*)((void*)(p)))

#if defined(__has_builtin)
#  if __has_builtin(__builtin_amdgcn_global_load_async_to_lds_b128)
#    define HAVE_ASYNC_LDS 1
#  else
#    define HAVE_ASYNC_LDS 0
#  endif
#  if __has_builtin(__builtin_amdgcn_cvt_pk_fp8_f32)
#    define HAVE_CVT_FP8 1
#  else
#    define HAVE_CVT_FP8 0
#  endif
#else
#  define HAVE_ASYNC_LDS 0
#  define HAVE_CVT_FP8 0
#endif

template <int N>
__device__ __forceinline__ void wait_async_le() {
#if defined(__has_builtin) && __has_builtin(__builtin_amdgcn_s_wait_asynccnt)
    __builtin_amdgcn_s_wait_asynccnt(N);
#else
    asm volatile("s_wait_asynccnt %0" ::"n"(N));
#endif
}

#if HAVE_ASYNC_LDS
#define ASYNC_B128(g, l, off)                                                  \
    __builtin_amdgcn_global_load_async_to_lds_b128(GLOBAL_AS(g), LDS_AS(l),    \
                                                   (off), 0)
#endif

// ---------------- math helpers ----------------
__device__ __forceinline__ float fast_silu(float v) {
    // v * sigmoid(v) = v / (1 + exp(-v)); exp via v_exp_f32 (base-2)
    float e = __builtin_amdgcn_exp2f(v * -1.44269504088896340736f);
    return v * __builtin_amdgcn_rcpf(1.0f + e);
}

__device__ __forceinline__ float clamp448(float v) {
    return fminf(fmaxf(v, -FP8_MAX), FP8_MAX);
}

#if !HAVE_CVT_FP8
// software float -> fp8 e4m3fn (RNE, clamp assumed done); fallback only
__device__ __forceinline__ unsigned sw_e4m3(float f) {
    union { float f; unsigned u; } v;
    v.f = f;
    unsigned sign = (v.u >> 31) << 7;
    float a = fabsf(f);
    if (a < 0.001953125f) return sign;  // below denorm range -> signed zero
    v.f = a;
    unsigned u = v.u;
    unsigned lsb = (u >> 20) & 1u;
    u += 0x7FFFFu + lsb;                // RNE at 3 mantissa bits
    int e8 = (int)(u >> 23) - 127 + 7;
    unsigned man = (u >> 20) & 7u;
    if (e8 <= 0) return sign;           // (approx) flush tiny denorms
    if (e8 > 15 || (e8 == 15 && man > 6)) { e8 = 15; man = 6; }  // clamp 448
    return sign | ((unsigned)e8 << 3) | man;
}
#endif

__device__ __forceinline__ unsigned pack4_fp8(float a, float b, float c, float d) {
#if HAVE_CVT_FP8
    int r = __builtin_amdgcn_cvt_pk_fp8_f32(a, b, 0, false);
    r     = __builtin_amdgcn_cvt_pk_fp8_f32(c, d, r, true);
    return (unsigned)r;
#else
    return sw_e4m3(a) | (sw_e4m3(b) << 8) | (sw_e4m3(c) << 16) |
           (sw_e4m3(d) << 24);
#endif
}

// compute 8 outputs from thread-private data, pack to 64 bits of fp8
__device__ __forceinline__ uint64_t swiglu8(const v4f a0, const v4f a1,
                                            const v4f b0, const v4f b1,
                                            float inv) {
    float y0 = clamp448(fast_silu(a0.x) * b0.x * inv);
    float y1 = clamp448(fast_silu(a0.y) * b0.y * inv);
    float y2 = clamp448(fast_silu(a0.z) * b0.z * inv);
    float y3 = clamp448(fast_silu(a0.w) * b0.w * inv);
    float y4 = clamp448(fast_silu(a1.x) * b1.x * inv);
    float y5 = clamp448(fast_silu(a1.y) * b1.y * inv);
    float y6 = clamp448(fast_silu(a1.z) * b1.z * inv);
    float y7 = clamp448(fast_silu(a1.w) * b1.w * inv);
    unsigned lo = pack4_fp8(y0, y1, y2, y3);
    unsigned hi = pack4_fp8(y4, y5, y6, y7);
    return (uint64_t)lo | ((uint64_t)hi << 32);
}

// ---------------- kernel ----------------
__global__ __launch_bounds__(THREADS) void swiglu_fp8_kernel(
    const float* __restrict__ x, const float* __restrict__ scale,
    uint8_t* __restrict__ out) {
    const int t   = threadIdx.x;
    const int bid = blockIdx.x;
    const float inv = __builtin_amdgcn_rcpf(scale[0]);

#if HAVE_ASYNC_LDS
    // DEPTH stages, each stage: 2048 floats of x1 + 2048 floats of x2 = 16KB
    __shared__ float lds[DEPTH * 2 * CHUNK];  // 64 KB

    // issue one chunk's 4 async B128 loads into stage s (thread-private slots)
    auto issue = [&](int k) {
        const int c = bid + k * GRID;
        const int s = k % DEPTH;
        const float* g1 = x + (size_t)(c >> 1) * COLS + (size_t)(c & 1) * CHUNK
                            + (size_t)t * PER_T;
        const float* g2 = g1 + HALF;
        float* l1 = lds + s * (2 * CHUNK) + t * PER_T;
        float* l2 = l1 + CHUNK;
        ASYNC_B128(g1, l1, 0);
        ASYNC_B128(g1, l1, 16);
        ASYNC_B128(g2, l2, 0);
        ASYNC_B128(g2, l2, 16);
    };

    int k_issue = 0;
#pragma unroll
    for (int s = 0; s < DEPTH; ++s) { issue(k_issue); ++k_issue; }

#pragma unroll
    for (int k = 0; k < CHUNKS_PER_BLOCK; ++k) {
        // exact in-flight accounting: 4 async ops per chunk, per wave
        const int inflight = k_issue - k;
        switch (inflight) {
        case 4: wait_async_le<12>(); break;
        case 3: wait_async_le<8>();  break;
        case 2: wait_async_le<4>();  break;
        default: wait_async_le<0>(); break;
        }

        const int c = bid + k * GRID;
        const float* l1 = lds + (k % DEPTH) * (2 * CHUNK) + t * PER_T;
        const float* l2 = l1 + CHUNK;
        v4f a0 = *(const v4f*)(l1);
        v4f a1 = *(const v4f*)(l1 + 4);
        v4f b0 = *(const v4f*)(l2);
        v4f b1 = *(const v4f*)(l2 + 4);

        uint64_t r = swiglu8(a0, a1, b0, b1, inv);
        uint64_t* po = (uint64_t*)(out + (size_t)c * CHUNK + (size_t)t * PER_T);
        __builtin_nontemporal_store(r, po);  // write-once output, keep L2 clean

        if (k_issue < CHUNKS_PER_BLOCK) { issue(k_issue); ++k_issue; }
    }
#else
    // fallback: direct vectorized nontemporal loads (no async path available)
#pragma unroll
    for (int k = 0; k < CHUNKS_PER_BLOCK; ++k) {
        const int c = bid + k * GRID;
        const float* g1 = x + (size_t)(c >> 1) * COLS + (size_t)(c & 1) * CHUNK
                            + (size_t)t * PER_T;
        const float* g2 = g1 + HALF;
        v4f a0 = __builtin_nontemporal_load((const v4f*)g1);
        v4f a1 = __builtin_nontemporal_load((const v4f*)(g1 + 4));
        v4f b0 = __builtin_nontemporal_load((const v4f*)g2);
        v4f b1 = __builtin_nontemporal_load((const v4f*)(g2 + 4));
        uint64_t r = swiglu8(a0, a1, b0, b1, inv);
        uint64_t* po = (uint64_t*)(out + (size_t)c * CHUNK + (size_t)t * PER_T);
        __builtin_nontemporal_store(r, po);
    }
#endif
}

// ---------------- launcher ----------------
extern "C" void kernel_launch(void* const* d_in, const int* in_sizes, int n_in,
                              void* d_out, int out_size, void* d_ws,
                              size_t ws_size, hipStream_t stream) {
    (void)in_sizes; (void)n_in; (void)d_ws; (void)ws_size; (void)out_size;
    const float* x     = (const float*)d_in[0];
    const float* scale = (const float*)d_in[1];
    uint8_t* out       = (uint8_t*)d_out;  // fp8 e4m3fn, 1 byte/element
    swiglu_fp8_kernel<<<GRID, THREADS, 0, stream>>>(x, scale, out);
}